// vq_codebook_6030134083833
// MI455X (gfx1250) — compile-verified
//
#include <hip/hip_runtime.h>

typedef __attribute__((ext_vector_type(2))) float v2f;
typedef __attribute__((ext_vector_type(4))) float v4f;
typedef __attribute__((ext_vector_type(8))) float v8f;

// One wave handles 16 rows of X per tile iteration.
// A (16x4, f32)  : lane l holds row l%16, K-pair 2*(l/16)          -> v2f
// B (4x16, f32)  : lane l holds col l%16, K-pair 2*(l/16)          -> v2f (per 16-col chunk)
// C/D (16x16 f32): VGPR r, lanes 0-15 -> row r, lanes 16-31 -> r+8 -> v8f
// C operand carries ||c||^2 (column-constant -> identical in all 8 VGPRs), so
// the WMMA directly produces dist^2 - ||x||^2 and no per-element add is needed.
__global__ __launch_bounds__(256) void vq_nn_kernel(
    const float* __restrict__ X, const float* __restrict__ tlut,
    float* __restrict__ hatX, int* __restrict__ state, int ntiles)
{
  const int lane  = threadIdx.x & 31;
  const int laneN = lane & 15;     // row index for A / col index for B,D
  const int half  = lane >> 4;     // 0 or 1
  const int khalf = half * 2;      // K offset within vec4: 0 or 2

  const int wavesPerBlock = blockDim.x >> 5;
  // Wave id is wave-uniform: force it scalar so the tile loop is SALU-controlled.
  const int gwave  = __builtin_amdgcn_readfirstlane(
      blockIdx.x * wavesPerBlock + (threadIdx.x >> 5));
  const int nwaves = gridDim.x * wavesPerBlock;

  // Loop-invariant: 16 chunks of B-matrix (16 codewords each) and the
  // broadcast ||c||^2 C-matrices (one v8f per chunk, all components equal).
  v2f bmat[16];
  v8f cmat[16];
#pragma unroll
  for (int j = 0; j < 16; ++j) {
    const float* cw = tlut + (j * 16 + laneN) * 4;
    v4f c4 = *(const v4f*)cw;
    float cs = c4.x * c4.x + c4.y * c4.y + c4.z * c4.z + c4.w * c4.w;
    v2f b;
    b.x = half ? c4.z : c4.x;
    b.y = half ? c4.w : c4.y;
    bmat[j] = b;
#pragma unroll
    for (int e = 0; e < 8; ++e) cmat[j][e] = cs;
  }

  if (gwave >= ntiles) return;

  // Software pipeline: A-tile for iteration t is loaded during iteration t-1.
  const int laneOff = laneN * 4 + khalf;
  v2f a_raw = *(const v2f*)(X + (size_t)(gwave << 4) + laneOff);

  for (int t = gwave; t < ntiles; t += nwaves) {
    // Issue next tile's load now; its s_wait lands after this tile's WMMAs.
    const int tn  = t + nwaves;
    const int tnc = (tn < ntiles) ? tn : t;          // clamped (uniform) address
    v2f a_next = *(const v2f*)(X + (size_t)(tnc << 4) + laneOff);

    v2f a;
    a.x = a_raw.x * -2.0f;                           // fold -2 of cross term into A
    a.y = a_raw.y * -2.0f;

    float bestV[8];
    int   bestI[8];

    // 16 chunks in groups of 4: 4 independent WMMAs in flight (covers the
    // WMMA->VALU hazard window) while capping live D registers at 32 VGPRs.
#pragma unroll
    for (int g = 0; g < 4; ++g) {
      v8f acc[4];
#pragma unroll
      for (int u = 0; u < 4; ++u) {
        acc[u] = __builtin_amdgcn_wmma_f32_16x16x4_f32(
            false, a, false, bmat[g * 4 + u], (short)0,
            cmat[g * 4 + u], false, false);
      }
#pragma unroll
      for (int u = 0; u < 4; ++u) {
        const int j    = g * 4 + u;
        const int cand = (j << 4) + laneN;           // codeword index this lane covers
        if (j == 0) {
#pragma unroll
          for (int r = 0; r < 8; ++r) { bestV[r] = acc[u][r]; bestI[r] = cand; }
        } else {
#pragma unroll
          for (int r = 0; r < 8; ++r) {
            bool take = acc[u][r] < bestV[r];        // strict '<': lowest index wins ties
            bestV[r]  = take ? acc[u][r] : bestV[r];
            bestI[r]  = take ? cand : bestI[r];
          }
        }
      }
      __builtin_amdgcn_sched_barrier(0);             // keep groups from merging
    }

    // Reduce across the 16 lanes of each half-wave (cols are lane-striped mod 16).
    // xor masks 1,2,4,8 never cross the 16-lane boundary on wave32.
#pragma unroll
    for (int r = 0; r < 8; ++r) {
#pragma unroll
      for (int s = 8; s >= 1; s >>= 1) {
        float ov  = __shfl_xor(bestV[r], s, 32);
        int   oi  = __shfl_xor(bestI[r], s, 32);
        bool take = ov < bestV[r];
        bestV[r]  = take ? ov : bestV[r];
        bestI[r]  = take ? oi : bestI[r];
      }
    }

    // Lanes 0-7 of each half write rows rowBase + (laneN + 8*half).
    if (laneN < 8) {
      int idx = bestI[0];
#pragma unroll
      for (int q = 1; q < 8; ++q) idx = (laneN == q) ? bestI[q] : idx;
      const int m = (t << 4) + laneN + (half << 3);
      state[m] = idx;
      *(v4f*)(hatX + (size_t)m * 4) = *(const v4f*)(tlut + (size_t)idx * 4);
    }

    a_raw = a_next;
  }
}

extern "C" void kernel_launch(void* const* d_in, const int* in_sizes, int n_in,
                              void* d_out, int out_size, void* d_ws, size_t ws_size,
                              hipStream_t stream) {
  const float* X    = (const float*)d_in[0];   // [B,4] f32
  const float* tlut = (const float*)d_in[1];   // [256,4] f32
  const int B       = in_sizes[0] / 4;
  const int ntiles  = B / 16;                  // B = 1M -> 65536 tiles

  float* hatX = (float*)d_out;                 // [B,4] f32
  int* state  = (int*)d_out + (size_t)B * 4;   // [B] i32, concatenated after hatX

  int blocks = 2048;                            // 16384 waves -> ~4 tiles/wave
  int maxBlocks = (ntiles + 7) / 8;             // >= 1 tile per wave
  if (blocks > maxBlocks) blocks = maxBlocks;
  if (blocks < 1) blocks = 1;

  vq_nn_kernel<<<blocks, 256, 0, stream>>>(X, tlut, hatX, state, ntiles);
}